// GMMConv_57612691309017
// MI455X (gfx1250) — compile-verified
//
#include <hip/hip_runtime.h>

typedef __attribute__((ext_vector_type(2))) float v2f;
typedef __attribute__((ext_vector_type(8))) float v8f;

#define F_CH 32
#define D_PS 3

__global__ void gmm_zero_kernel(float4* __restrict__ acc, int n4) {
    int i = blockIdx.x * blockDim.x + threadIdx.x;
    if (i < n4) acc[i] = make_float4(0.f, 0.f, 0.f, 0.f);
}

// One wave (32 lanes) per edge; lane = feature channel.
// acc[row][f] += x[col][f] * exp2(sum_d -(p_d - mu_fd)^2 * log2e / (2 sigma_fd^2))
__global__ void gmm_edge_kernel(const float* __restrict__ x,
                                const long long* __restrict__ ei_row,
                                const long long* __restrict__ ei_col,
                                const float* __restrict__ pseudo,
                                const float* __restrict__ mu,
                                const float* __restrict__ sigma,
                                float* __restrict__ acc,
                                int E) {
    __shared__ float s_mu[F_CH * D_PS];
    __shared__ float s_inv[F_CH * D_PS];
    int t = threadIdx.x;
    if (t < F_CH * D_PS) {
        s_mu[t] = mu[t];
        float s = sigma[t];
        // fold -0.5 and log2(e) so the weight is a single v_exp_f32 (exp2)
        s_inv[t] = -0.72134752f / (1e-14f + s * s);  // -0.5 * log2(e)
    }
    __syncthreads();

    int gid = blockIdx.x * blockDim.x + threadIdx.x;
    int e = gid >> 5;          // edge index (wave-uniform)
    int f = gid & 31;          // channel = lane
    if (e >= E) return;
    e = __builtin_amdgcn_readfirstlane(e);   // make uniformity explicit -> s_load path

    float p0 = pseudo[e * 3 + 0];
    float p1 = pseudo[e * 3 + 1];
    float p2 = pseudo[e * 3 + 2];
    float d0 = p0 - s_mu[f * 3 + 0];
    float d1 = p1 - s_mu[f * 3 + 1];
    float d2 = p2 - s_mu[f * 3 + 2];
    float l2w = d0 * d0 * s_inv[f * 3 + 0]
              + d1 * d1 * s_inv[f * 3 + 1]
              + d2 * d2 * s_inv[f * 3 + 2];
    float w = exp2f(l2w);

    long long c = ei_col[e];   // scalar
    long long r = ei_row[e];   // scalar
    float msg = x[c * F_CH + f] * w;

    // Native non-returning fp32 atomic at device scope (RMW at L2, STOREcnt).
    float* p = acc + r * F_CH + f;
    asm volatile("global_atomic_add_f32 %0, %1, off scope:SCOPE_DEV"
                 :: "v"(p), "v"(msg) : "memory");
}

// out(N,32) = acc(N,32) @ W^T(32,32) + b, via V_WMMA_F32_16X16X4_F32.
// One wave per 16-row tile; K=32 in 8 steps; two N-tiles of 16; bias pre-loaded into C.
__global__ void gmm_linear_wmma_kernel(const float* __restrict__ acc,
                                       const float* __restrict__ W,   // (O=32, F=32)
                                       const float* __restrict__ bias,
                                       float* __restrict__ out,
                                       int N, int ntiles) {
    int lane = threadIdx.x & 31;
    int wave = threadIdx.x >> 5;
    int tile = blockIdx.x * (blockDim.x >> 5) + wave;
    if (tile >= ntiles) return;               // wave-uniform: EXEC stays all-ones

    bool fulltile = (tile * 16 + 16 <= N);    // wave-uniform
    int mrow = tile * 16 + (lane & 15);
    if (mrow > N - 1) mrow = N - 1;           // clamp loads on ragged tail
    int khalf = (lane >> 4) * 2;              // lanes 0-15 -> K{+0,+1}; 16-31 -> K{+2,+3}
    int nlo = lane & 15;
    const float* arow = acc + (long long)mrow * F_CH;

    for (int nt = 0; nt < 2; ++nt) {
        int n = nt * 16 + nlo;                // output channel for this lane
        const float* wrow = W + n * F_CH;     // B[k][n] = W[n][k]
        float bv = bias[n];
        v8f c;
#pragma unroll
        for (int r = 0; r < 8; ++r) c[r] = bv;   // bias folded into accumulator init
#pragma unroll
        for (int k = 0; k < 8; ++k) {
            int kk = k * 4 + khalf;
            v2f a;  a.x = arow[kk];  a.y = arow[kk + 1];
            v2f bb; bb.x = wrow[kk]; bb.y = wrow[kk + 1];
            c = __builtin_amdgcn_wmma_f32_16x16x4_f32(
                    false, a, false, bb, (short)0, c, false, false);
        }
        int mbase = tile * 16 + (lane >> 4) * 8;  // C/D: VGPR r -> M = mbase + r
        float* op = out + (long long)mbase * F_CH + n;
        if (fulltile) {
#pragma unroll
            for (int r = 0; r < 8; ++r) op[(long long)r * F_CH] = c[r];
        } else {
#pragma unroll
            for (int r = 0; r < 8; ++r)
                if (mbase + r < N) op[(long long)r * F_CH] = c[r];
        }
    }
}

extern "C" void kernel_launch(void* const* d_in, const int* in_sizes, int n_in,
                              void* d_out, int out_size, void* d_ws, size_t ws_size,
                              hipStream_t stream) {
    const float*     x      = (const float*)d_in[0];
    const long long* ei     = (const long long*)d_in[1]; // int64 (2,E): row then col
    const float*     pseudo = (const float*)d_in[2];
    const float*     mu     = (const float*)d_in[3];
    const float*     sigma  = (const float*)d_in[4];
    const float*     W      = (const float*)d_in[5];
    const float*     b      = (const float*)d_in[6];
    float*           out    = (float*)d_out;

    int N = in_sizes[0] / F_CH;       // 100000
    int E = in_sizes[1] / 2;          // 1600000

    float* acc = (float*)d_ws;        // N*32 floats = 12.8 MB scratch
    int nacc = N * F_CH;
    int n4 = nacc / 4;                // N*32 divisible by 4

    gmm_zero_kernel<<<(n4 + 255) / 256, 256, 0, stream>>>((float4*)acc, n4);

    long long tot = (long long)E * 32;
    int eblocks = (int)((tot + 255) / 256);
    gmm_edge_kernel<<<eblocks, 256, 0, stream>>>(x, ei, ei + E, pseudo, mu, sigma, acc, E);

    int ntiles = (N + 15) / 16;
    gmm_linear_wmma_kernel<<<(ntiles + 7) / 8, 256, 0, stream>>>(acc, W, b, out, N, ntiles);
}